// GeometricAttentionLayer_89567247990900
// MI455X (gfx1250) — compile-verified
//
#include <hip/hip_runtime.h>
#include <math.h>

typedef __attribute__((ext_vector_type(2))) float v2f;
typedef __attribute__((ext_vector_type(8))) float v8f;

// ---------------------------------------------------------------------------
// Fused QKV projection: {q,k,v}[M,128] = X[M,256] @ {Wq,Wk,Wv}^T + bias
// One block (8 waves, 256 thr) per 16-row M-tile; x-tile staged in LDS once.
// Unrolled t-loop: t = matrix (compile-time), wave = output column tile, so
// W/bias/Y pointers stay in the global address space (global_load_b64 path).
// ---------------------------------------------------------------------------
__global__ __launch_bounds__(256) void wmma_qkv_f32(
    const float* __restrict__ X,
    const float* __restrict__ Wq, const float* __restrict__ bq,
    const float* __restrict__ Wk, const float* __restrict__ bk,
    const float* __restrict__ Wv, const float* __restrict__ bv,
    float* __restrict__ qo, float* __restrict__ ko, float* __restrict__ vo,
    int M)
{
    const int K = 256, OUTD = 128;
    __shared__ float lds[16 * 256];   // 16 KB x-tile

    const int tid  = threadIdx.x;
    const int wave = tid >> 5;        // 0..7  -> output column tile
    const int lane = tid & 31;
    const int half = lane >> 4;
    const int l15  = lane & 15;
    const int m0   = blockIdx.x * 16;

    // cooperative float4 load of x tile [16 x 256]
    for (int i = tid; i < (16 * K) >> 2; i += 256) {
        const int idx = i << 2;
        const int r = idx >> 8;           // / 256
        const int c = idx & 255;
        float4 val = make_float4(0.f, 0.f, 0.f, 0.f);
        if (m0 + r < M)
            val = *(const float4*)(X + (size_t)(m0 + r) * K + c);
        *(float4*)(&lds[r * K + c]) = val;
    }
    __syncthreads();

    const int    n0   = wave * 16;                 // output column tile
    const size_t woff = (size_t)(n0 + l15) * K;    // W row for this lane

    #pragma unroll
    for (int t = 0; t < 3; ++t) {
        const float* W  = (t == 0) ? Wq : (t == 1) ? Wk : Wv;   // compile-time
        const float* bb = (t == 0) ? bq : (t == 1) ? bk : bv;
        float*       Y  = (t == 0) ? qo : (t == 1) ? ko : vo;

        const float* wrow = W + woff;
        v8f acc = {0.f, 0.f, 0.f, 0.f, 0.f, 0.f, 0.f, 0.f};
        for (int k = 0; k < K; k += 4) {
            const int ka = k + half * 2;
            v2f a, b;
            a.x = lds[l15 * K + ka];
            a.y = lds[l15 * K + ka + 1];
            b.x = wrow[ka];
            b.y = wrow[ka + 1];
            acc = __builtin_amdgcn_wmma_f32_16x16x4_f32(
                false, a, false, b, (short)0, acc, false, false);
        }
        const float bv_ = bb[n0 + l15];
        #pragma unroll
        for (int r = 0; r < 8; ++r) {
            const int row = m0 + r + half * 8;
            if (row < M)
                Y[(size_t)row * OUTD + n0 + l15] = acc[r] + bv_;
        }
    }
}

// ---------------------------------------------------------------------------
// Generic Y[M,OUTD] = X[M,K] @ W[OUTD,K]^T + bias  (output projection)
// 8 waves / block, one 16x16 tile per wave, grid = (ceil(M/16), OUTD/128)
// ---------------------------------------------------------------------------
__global__ __launch_bounds__(256) void wmma_linear_f32(
    const float* __restrict__ X, const float* __restrict__ W,
    const float* __restrict__ bias, float* __restrict__ Y,
    int M, int K, int OUTD)
{
    __shared__ float lds[16 * 256];   // max K = 256

    const int tid  = threadIdx.x;
    const int wave = tid >> 5;
    const int lane = tid & 31;
    const int half = lane >> 4;
    const int l15  = lane & 15;
    const int m0   = blockIdx.x * 16;

    for (int i = tid; i < (16 * K) >> 2; i += 256) {
        const int idx = i << 2;
        const int r = idx / K;
        const int c = idx - r * K;
        float4 val = make_float4(0.f, 0.f, 0.f, 0.f);
        if (m0 + r < M)
            val = *(const float4*)(X + (size_t)(m0 + r) * K + c);
        *(float4*)(&lds[r * K + c]) = val;
    }
    __syncthreads();

    const int n0 = (blockIdx.y * 8 + wave) * 16;
    if (n0 < OUTD) {
        const float* wrow = W + (size_t)(n0 + l15) * K;
        v8f acc = {0.f, 0.f, 0.f, 0.f, 0.f, 0.f, 0.f, 0.f};
        for (int k = 0; k < K; k += 4) {
            const int ka = k + half * 2;
            v2f a, b;
            a.x = lds[l15 * K + ka];
            a.y = lds[l15 * K + ka + 1];
            b.x = wrow[ka];
            b.y = wrow[ka + 1];
            acc = __builtin_amdgcn_wmma_f32_16x16x4_f32(
                false, a, false, b, (short)0, acc, false, false);
        }
        const float bv = bias[n0 + l15];
        #pragma unroll
        for (int r = 0; r < 8; ++r) {
            const int row = m0 + r + half * 8;
            if (row < M)
                Y[(size_t)row * OUTD + n0 + l15] = acc[r] + bv;
        }
    }
}

// ---------------------------------------------------------------------------
__global__ void init_ws_kernel(float* __restrict__ m, float* __restrict__ s,
                               float* __restrict__ acc, int N)
{
    const int t = blockIdx.x * blockDim.x + threadIdx.x;
    if (t < N * 8) { m[t] = -INFINITY; s[t] = 0.f; }
    if (t < N * 128) acc[t] = 0.f;
}

// sign-aware float atomic max (valid with -inf init)
__device__ inline void atomicMaxFloat(float* addr, float val)
{
    if (val >= 0.f)
        atomicMax((int*)addr, __float_as_int(val));
    else
        atomicMin((unsigned int*)addr, __float_as_uint(val));
}

// ---------------------------------------------------------------------------
// per-edge: RBF -> geo MLP -> attention logits ; atomic segment-max
// ---------------------------------------------------------------------------
__global__ __launch_bounds__(256) void edge_logits_kernel(
    const float* __restrict__ q, const float* __restrict__ kmat,
    const int* __restrict__ eidx, const float* __restrict__ dist,
    const float* __restrict__ Wg1, const float* __restrict__ bg1,
    const float* __restrict__ Wg2, const float* __restrict__ bg2,
    float* __restrict__ attn, float* __restrict__ mmax, int E)
{
    __shared__ float sWg1[256], sbg1[16], sWg2[128], sbg2[8];
    for (int i = threadIdx.x; i < 256; i += 256) sWg1[i] = Wg1[i];
    for (int i = threadIdx.x; i < 128; i += 256) sWg2[i] = Wg2[i];
    if (threadIdx.x < 16) sbg1[threadIdx.x] = bg1[threadIdx.x];
    if (threadIdx.x < 8)  sbg2[threadIdx.x] = bg2[threadIdx.x];
    __syncthreads();

    const int e = blockIdx.x * blockDim.x + threadIdx.x;
    if (e >= E) return;

    // RBF: 16 centers linspace(0,6) -> step 0.4, gamma = 1
    const float d = dist[e];
    float rbf[16];
    #pragma unroll
    for (int i = 0; i < 16; ++i) {
        const float t = d - 0.4f * (float)i;
        rbf[i] = expf(-t * t);
    }
    float hid[16];
    #pragma unroll
    for (int i = 0; i < 16; ++i) {
        float a = sbg1[i];
        #pragma unroll
        for (int j = 0; j < 16; ++j) a = fmaf(sWg1[i * 16 + j], rbf[j], a);
        hid[i] = fmaxf(a, 0.f);
    }

    const int row = eidx[e];
    const int col = eidx[E + e];
    const float4* qr = (const float4*)(q    + (size_t)row * 128);
    const float4* kc = (const float4*)(kmat + (size_t)col * 128);
    const float scale = 0.25f;   // 1/sqrt(16)

    #pragma unroll
    for (int h = 0; h < 8; ++h) {
        float gb = sbg2[h];
        #pragma unroll
        for (int j = 0; j < 16; ++j) gb = fmaf(sWg2[h * 16 + j], hid[j], gb);
        float dot = 0.f;
        #pragma unroll
        for (int j4 = 0; j4 < 4; ++j4) {
            const float4 qv = qr[h * 4 + j4];
            const float4 kv = kc[h * 4 + j4];
            dot = fmaf(qv.x, kv.x, dot);
            dot = fmaf(qv.y, kv.y, dot);
            dot = fmaf(qv.z, kv.z, dot);
            dot = fmaf(qv.w, kv.w, dot);
        }
        const float a = dot * scale + gb;
        attn[(size_t)e * 8 + h] = a;
        atomicMaxFloat(&mmax[(size_t)row * 8 + h], a);
    }
}

// ---------------------------------------------------------------------------
// per-edge: e = exp(attn - m[row]) (in place) ; atomic segment-sum
// ---------------------------------------------------------------------------
__global__ __launch_bounds__(256) void edge_exp_kernel(
    const int* __restrict__ eidx, float* __restrict__ attn,
    const float* __restrict__ mmax, float* __restrict__ ssum, int E)
{
    const int e = blockIdx.x * blockDim.x + threadIdx.x;
    if (e >= E) return;
    const int row = eidx[e];
    #pragma unroll
    for (int h = 0; h < 8; ++h) {
        const float v = expf(attn[(size_t)e * 8 + h] - mmax[(size_t)row * 8 + h]);
        attn[(size_t)e * 8 + h] = v;
        atomicAdd(&ssum[(size_t)row * 8 + h], v);
    }
}

// ---------------------------------------------------------------------------
// per-(edge,head): normalized scatter-add of v[col] into acc[row]
// ---------------------------------------------------------------------------
__global__ __launch_bounds__(256) void edge_agg_kernel(
    const int* __restrict__ eidx, const float* __restrict__ attn,
    const float* __restrict__ ssum, const float* __restrict__ v,
    float* __restrict__ acc, int E)
{
    const int t = blockIdx.x * blockDim.x + threadIdx.x;
    const int e = t >> 3;
    const int h = t & 7;
    if (e >= E) return;
    const int row = eidx[e];
    const int col = eidx[E + e];
    const float a = attn[(size_t)e * 8 + h] /
                    (ssum[(size_t)row * 8 + h] + 1e-12f);
    const float4* vp = (const float4*)(v + (size_t)col * 128 + h * 16);
    float*        op = acc + (size_t)row * 128 + h * 16;
    #pragma unroll
    for (int j4 = 0; j4 < 4; ++j4) {
        const float4 vv = vp[j4];
        atomicAdd(op + j4 * 4 + 0, vv.x * a);
        atomicAdd(op + j4 * 4 + 1, vv.y * a);
        atomicAdd(op + j4 * 4 + 2, vv.z * a);
        atomicAdd(op + j4 * 4 + 3, vv.w * a);
    }
}

// ---------------------------------------------------------------------------
extern "C" void kernel_launch(void* const* d_in, const int* in_sizes, int n_in,
                              void* d_out, int out_size, void* d_ws, size_t ws_size,
                              hipStream_t stream)
{
    const float* x    = (const float*)d_in[0];
    const int*   eidx = (const int*)  d_in[1];   // [2,E]: row=eidx[e], col=eidx[E+e]
    const float* dist = (const float*)d_in[2];
    const float* Wq   = (const float*)d_in[3];
    const float* bq   = (const float*)d_in[4];
    const float* Wk   = (const float*)d_in[5];
    const float* bk   = (const float*)d_in[6];
    const float* Wv   = (const float*)d_in[7];
    const float* bv   = (const float*)d_in[8];
    const float* Wo   = (const float*)d_in[9];
    const float* bo   = (const float*)d_in[10];
    const float* Wg1  = (const float*)d_in[11];
    const float* bg1  = (const float*)d_in[12];
    const float* Wg2  = (const float*)d_in[13];
    const float* bg2  = (const float*)d_in[14];

    const int N = in_sizes[0] / 256;   // 50000
    const int E = in_sizes[2];         // 800000

    float* q    = (float*)d_ws;
    float* k    = q    + (size_t)N * 128;
    float* v    = k    + (size_t)N * 128;
    float* attn = v    + (size_t)N * 128;
    float* m    = attn + (size_t)E * 8;
    float* s    = m    + (size_t)N * 8;
    float* acc  = s    + (size_t)N * 8;

    const int Mtiles = (N + 15) / 16;

    // 1) init m/s/acc
    init_ws_kernel<<<(N * 128 + 255) / 256, 256, 0, stream>>>(m, s, acc, N);

    // 2) fused q/k/v projection (x staged in LDS once per M-tile)
    wmma_qkv_f32<<<Mtiles, 256, 0, stream>>>(x, Wq, bq, Wk, bk, Wv, bv,
                                             q, k, v, N);

    // 3) edge logits + segment max
    edge_logits_kernel<<<(E + 255) / 256, 256, 0, stream>>>(
        q, k, eidx, dist, Wg1, bg1, Wg2, bg2, attn, m, E);

    // 4) exp + segment sum
    edge_exp_kernel<<<(E + 255) / 256, 256, 0, stream>>>(eidx, attn, m, s, E);

    // 5) normalized aggregation
    {
        const long long total = (long long)E * 8;
        edge_agg_kernel<<<(int)((total + 255) / 256), 256, 0, stream>>>(
            eidx, attn, s, v, acc, E);
    }

    // 6) output projection
    {
        dim3 grid(Mtiles, 1);
        wmma_linear_f32<<<grid, 256, 0, stream>>>(acc, Wo, bo, (float*)d_out,
                                                  N, 128, 128);
    }
}